// PostProcessor_36910948942013
// MI455X (gfx1250) — compile-verified
//
#include <hip/hip_runtime.h>
#include <math.h>

// ---------------------------------------------------------------------------
// Problem constants (must match reference)
// ---------------------------------------------------------------------------
#define N_PROP      600
#define NUM_CLASSES 151
#define FEAT_DIM    4096
#define DETS        100
#define TOPN_CLS    300
#define SCORE_TH    0.05f
#define NMS_TH      0.5f
#define IMG_W       1024.0f
#define IMG_H       768.0f
#define BBOX_CLIP   4.135166556742356f   // log(1000/16)
#define NEG_BIG     (-1e30f)

#define NMS_BLOCK   640                  // 20 wave32s, >= N_PROP threads
#define NW          19                   // bit-words covering 600 boxes
#define MW          20                   // padded row stride (words)

// ---------------------------------------------------------------------------
// gfx1250 async global<->LDS path (verified present on this toolchain).
// Prototype (from clang diagnostic): (AS1 int4*, AS3 int4*, imm off, imm cpol)
// ---------------------------------------------------------------------------
#ifndef __has_builtin
#define __has_builtin(x) 0
#endif

#if __has_builtin(__builtin_amdgcn_global_load_async_to_lds_b128) && \
    __has_builtin(__builtin_amdgcn_global_store_async_from_lds_b128)
#define HAVE_ASYNC_LDS 1
#else
#define HAVE_ASYNC_LDS 0
#endif

#if __has_builtin(__builtin_amdgcn_s_wait_asynccnt)
#define WAIT_ASYNCCNT_0() __builtin_amdgcn_s_wait_asynccnt(0)
#else
#define WAIT_ASYNCCNT_0() asm volatile("s_wait_asynccnt 0" ::: "memory")
#endif

typedef int v4i __attribute__((ext_vector_type(4)));
typedef __attribute__((address_space(1))) v4i global_v4i;   // AS1 (= __device__)
typedef __attribute__((address_space(3))) v4i shared_v4i;   // AS3 (= __shared__)

// ---------------------------------------------------------------------------
// Kernel A: softmax over classes + box decode + clip.  One wave32 / proposal.
// ---------------------------------------------------------------------------
__global__ __launch_bounds__(128) void k_softmax_decode(
    const float* __restrict__ logits, const float* __restrict__ reg,
    const float* __restrict__ pb, float* __restrict__ probs,
    float* __restrict__ boxes)
{
  const int p    = (blockIdx.x << 2) + (threadIdx.x >> 5);
  const int lane = threadIdx.x & 31;
  if (p >= N_PROP) return;

  const float* L = logits + (size_t)p * NUM_CLASSES;
  float mx = NEG_BIG;
  for (int c = lane; c < NUM_CLASSES; c += 32) mx = fmaxf(mx, L[c]);
  for (int o = 16; o; o >>= 1) mx = fmaxf(mx, __shfl_xor(mx, o, 32));
  float se = 0.0f;
  for (int c = lane; c < NUM_CLASSES; c += 32) se += expf(L[c] - mx);
  for (int o = 16; o; o >>= 1) se += __shfl_xor(se, o, 32);
  const float inv = 1.0f / se;
  float* P = probs + (size_t)p * NUM_CLASSES;
  for (int c = lane; c < NUM_CLASSES; c += 32) P[c] = expf(L[c] - mx) * inv;

  const float x1 = pb[p * 4 + 0], y1 = pb[p * 4 + 1];
  const float x2 = pb[p * 4 + 2], y2 = pb[p * 4 + 3];
  const float w  = x2 - x1 + 1.0f, h = y2 - y1 + 1.0f;
  const float cx = x1 + 0.5f * w,  cy = y1 + 0.5f * h;
  const float* R = reg   + (size_t)p * NUM_CLASSES * 4;
  float*       B = boxes + (size_t)p * NUM_CLASSES * 4;
  for (int c = lane; c < NUM_CLASSES; c += 32) {
    const float dx = R[c * 4 + 0] / 10.0f;
    const float dy = R[c * 4 + 1] / 10.0f;
    const float dw = fminf(R[c * 4 + 2] / 5.0f, BBOX_CLIP);
    const float dh = fminf(R[c * 4 + 3] / 5.0f, BBOX_CLIP);
    const float pcx = dx * w + cx, pcy = dy * h + cy;
    const float pw  = expf(dw) * w, ph = expf(dh) * h;
    B[c * 4 + 0] = fminf(fmaxf(pcx - 0.5f * pw, 0.0f), IMG_W - 1.0f);
    B[c * 4 + 1] = fminf(fmaxf(pcy - 0.5f * ph, 0.0f), IMG_H - 1.0f);
    B[c * 4 + 2] = fminf(fmaxf(pcx + 0.5f * pw - 1.0f, 0.0f), IMG_W - 1.0f);
    B[c * 4 + 3] = fminf(fmaxf(pcy + 0.5f * ph - 1.0f, 0.0f), IMG_H - 1.0f);
  }
}

// ---------------------------------------------------------------------------
// Kernel B: per-class bitmask NMS.  One workgroup per class.
//  1) stable-descending rank sort (O(N^2), matches jnp.argsort(-s) stable)
//  2) all 640 threads build the upper-triangular suppression bit-matrix
//     (bit j of word w in row i  <=>  j>i and IoU(i,j)>0.5)       [1 barrier]
//  3) wave 0 alone runs the 600-step greedy chain on 19 alive-words:
//     shfl broadcast + andn per step — no block barriers on the chain.
// ---------------------------------------------------------------------------
__global__ __launch_bounds__(NMS_BLOCK) void k_nms_per_class(
    const float* __restrict__ probs, const float* __restrict__ boxes,
    int* __restrict__ kept)
{
  __shared__ float        ss[N_PROP];
  __shared__ float        sbox[N_PROP * 4];
  __shared__ int          ord[N_PROP];
  __shared__ int          vs[N_PROP];
  __shared__ unsigned int mask[N_PROP * MW];   // 48 KB

  const int cls = blockIdx.x + 1;       // classes 1..150
  const int t   = threadIdx.x;

  if (t < N_PROP) {
    const float p = probs[(size_t)t * NUM_CLASSES + cls];
    ss[t] = (p > SCORE_TH) ? p : NEG_BIG;
  }
  __syncthreads();

  if (t < N_PROP) {
    const float sj = ss[t];
    int r = 0;
    for (int k = 0; k < N_PROP; ++k) {
      const float sk = ss[k];
      r += (sk > sj) || ((sk == sj) && (k < t));
    }
    ord[r] = t;
    vs[r]  = (sj > -1e29f) ? 1 : 0;
    const float* b = boxes + ((size_t)t * NUM_CLASSES + cls) * 4;
    sbox[r * 4 + 0] = b[0];
    sbox[r * 4 + 1] = b[1];
    sbox[r * 4 + 2] = b[2];
    sbox[r * 4 + 3] = b[3];
  }
  __syncthreads();

  // ---- phase: suppression bit-matrix (rows in sorted order) ----
  for (int task = t; task < N_PROP * NW; task += NMS_BLOCK) {
    const int i = task / NW;
    const int w = task % NW;
    unsigned int bits = 0u;
    if (w * 32 + 31 > i) {                 // word intersects the j>i triangle
      const float ax1 = sbox[i * 4 + 0], ay1 = sbox[i * 4 + 1];
      const float ax2 = sbox[i * 4 + 2], ay2 = sbox[i * 4 + 3];
      const float aA  = (ax2 - ax1 + 1.0f) * (ay2 - ay1 + 1.0f);
      const int jend  = min(w * 32 + 32, N_PROP);
      for (int j = max(w * 32, i + 1); j < jend; ++j) {
        const float bx1 = sbox[j * 4 + 0], by1 = sbox[j * 4 + 1];
        const float bx2 = sbox[j * 4 + 2], by2 = sbox[j * 4 + 3];
        const float bA  = (bx2 - bx1 + 1.0f) * (by2 - by1 + 1.0f);
        const float iw  = fmaxf(fminf(ax2, bx2) - fmaxf(ax1, bx1) + 1.0f, 0.0f);
        const float ih  = fmaxf(fminf(ay2, by2) - fmaxf(ay1, by1) + 1.0f, 0.0f);
        const float inter = iw * ih;
        const float iou   = inter / (aA + bA - inter);
        if (iou > NMS_TH) bits |= (1u << (j & 31));
      }
    }
    mask[i * MW + w] = bits;
  }
  __syncthreads();

  // ---- phase: greedy chain on wave 0 only (no block barriers) ----
  if (t < 32) {
    const int l = t;                      // lane owns alive/keep word l
    unsigned int alive = 0u;
    if (l < NW) {
      const int jb = l * 32;
      for (int b = 0; b < 32; ++b) {
        const int j = jb + b;
        if (j < N_PROP && vs[j]) alive |= (1u << b);
      }
    }
    unsigned int keepw = 0u;
    for (int i = 0; i < N_PROP; ++i) {
      const int wi = i >> 5, bi = i & 31;
      const unsigned int aw = __shfl(alive, wi, 32);   // broadcast word wi
      if ((aw >> bi) & 1u) {                           // box i survives -> keep
        if (l == wi) keepw |= (1u << bi);
        if (l < NW) alive &= ~mask[i * MW + l];        // kill overlapped j>i
      }
    }
    // top-300 cap: exclusive popcount prefix across lanes
    int base = 0;
    for (int k = 0; k < NW; ++k) {
      const unsigned int kw = __shfl(keepw, k, 32);
      if (k < l) base += __popc(kw);
    }
    if (l < NW) {
      for (int b = 0; b < 32; ++b) {
        const int j = l * 32 + b;
        if (j < N_PROP) {
          const int c  = base + __popc(keepw & ((2u << b) - 1u));  // inclusive
          const int kp = ((keepw >> b) & 1u) && (c <= TOPN_CLS);
          kept[(size_t)cls * N_PROP + ord[j]] = kp;
        }
      }
    }
  }
}

// ---------------------------------------------------------------------------
// Kernel C: nms_filter_duplicates -> per-proposal (score, label), sel.
// ---------------------------------------------------------------------------
__global__ __launch_bounds__(128) void k_label_select(
    const float* __restrict__ probs, const int* __restrict__ kept,
    float* __restrict__ sel, int* __restrict__ labels_pre)
{
  const int n = blockIdx.x * blockDim.x + threadIdx.x;
  if (n >= N_PROP) return;
  float best = 0.0f;
  int   bl   = 0;
  for (int c = 1; c < NUM_CLASSES; ++c) {
    if (kept[(size_t)c * N_PROP + n]) {
      const float v = probs[(size_t)n * NUM_CLASSES + c];
      if (v > best) { best = v; bl = c; }   // strict > keeps lowest class on ties
    }
  }
  sel[n]        = (best > 0.0f) ? best : -1.0f;
  labels_pre[n] = bl;
}

// ---------------------------------------------------------------------------
// Kernel D: top-100 (stable tie order == lax.top_k), emit scores/labels/valid/
// boxes, stash top_idx/top_m, prefetch feature rows into L2.
// ---------------------------------------------------------------------------
__global__ __launch_bounds__(NMS_BLOCK) void k_topk(
    const float* __restrict__ sel, const int* __restrict__ labels_pre,
    const float* __restrict__ boxes, const float* __restrict__ features,
    float* __restrict__ out_boxes, float* __restrict__ out_scores,
    float* __restrict__ out_labels, float* __restrict__ out_valid,
    int* __restrict__ top_idx, int* __restrict__ top_m)
{
  __shared__ float s[N_PROP];
  const int j = threadIdx.x;
  if (j < N_PROP) s[j] = sel[j];
  __syncthreads();
  if (j >= N_PROP) return;

  const float sj = s[j];
  int r = 0;
  for (int k = 0; k < N_PROP; ++k) {
    const float sk = s[k];
    r += (sk > sj) || ((sk == sj) && (k < j));
  }
  if (r >= DETS) return;

  const int   valid = (sj > 0.0f) ? 1 : 0;
  const float m     = valid ? 1.0f : 0.0f;
  const int   lbl   = labels_pre[j];
  out_scores[r] = valid ? sj : 0.0f;
  out_labels[r] = valid ? (float)lbl : 0.0f;
  out_valid[r]  = m;
  const float* b = boxes + ((size_t)j * NUM_CLASSES + lbl) * 4;
  out_boxes[r * 4 + 0] = b[0] * m;
  out_boxes[r * 4 + 1] = b[1] * m;
  out_boxes[r * 4 + 2] = b[2] * m;
  out_boxes[r * 4 + 3] = b[3] * m;
  top_idx[r] = j;
  top_m[r]   = valid;
  // warm L2 for the feature-gather kernel (global_prefetch_b8 on gfx1250)
  __builtin_prefetch(features + (size_t)j * FEAT_DIM, 0, 3);
}

// ---------------------------------------------------------------------------
// Kernel E: boxes_per_cls gather (100 x 151 x 4).
// ---------------------------------------------------------------------------
__global__ __launch_bounds__(256) void k_boxes_per_cls(
    const float* __restrict__ boxes, const int* __restrict__ top_idx,
    const int* __restrict__ top_m, float* __restrict__ outb)
{
  const int   r   = blockIdx.x;
  const int   idx = top_idx[r];
  const float m   = (float)top_m[r];
  const float* src = boxes + (size_t)idx * NUM_CLASSES * 4;
  float*       dst = outb  + (size_t)r   * NUM_CLASSES * 4;
  for (int e = threadIdx.x; e < NUM_CLASSES * 4; e += blockDim.x)
    dst[e] = src[e] * m;
}

// ---------------------------------------------------------------------------
// Kernel F: feature gather (100 rows x 16 KB) — VGPR-free async streaming via
// LDS on gfx1250 (global_load_async_to_lds_b128 / global_store_async_from_lds
// tracked by ASYNCcnt), plain float4 copy fallback otherwise.
// ---------------------------------------------------------------------------
__global__ __launch_bounds__(256) void k_feat_gather(
    const float* __restrict__ features, const int* __restrict__ top_idx,
    const int* __restrict__ top_m, float* __restrict__ outf)
{
  const int r   = blockIdx.x;
  const int idx = top_idx[r];       // always a valid proposal index
  const int m   = top_m[r];         // block-uniform: no divergence
  const int t   = threadIdx.x;
  float* dst = outf + (size_t)r * FEAT_DIM;

#if HAVE_ASYNC_LDS
  __shared__ __align__(16) unsigned char buf[FEAT_DIM * 4];
  if (m) {
    const char* g  = (const char*)(features + (size_t)idx * FEAT_DIM);
    char*       dg = (char*)dst;
#pragma unroll
    for (int it = 0; it < 4; ++it) {
      const int off = (it * 256 + t) * 16;
      __builtin_amdgcn_global_load_async_to_lds_b128(
          (global_v4i*)(g + off), (shared_v4i*)(buf + off), 0, 0);
    }
    WAIT_ASYNCCNT_0();              // load->LDS complete (wave-local regions)
#pragma unroll
    for (int it = 0; it < 4; ++it) {
      const int off = (it * 256 + t) * 16;
      __builtin_amdgcn_global_store_async_from_lds_b128(
          (global_v4i*)(dg + off), (shared_v4i*)(buf + off), 0, 0);
    }
    WAIT_ASYNCCNT_0();
  } else {
    float4* d4 = (float4*)dst;
    const float4 z = make_float4(0.f, 0.f, 0.f, 0.f);
#pragma unroll
    for (int it = 0; it < 4; ++it) d4[it * 256 + t] = z;
  }
#else
  const float4* s4 = (const float4*)(features + (size_t)idx * FEAT_DIM);
  float4*       d4 = (float4*)dst;
  const float   fm = (float)m;
#pragma unroll
  for (int it = 0; it < 4; ++it) {
    const float4 v = s4[it * 256 + t];
    d4[it * 256 + t] = make_float4(v.x * fm, v.y * fm, v.z * fm, v.w * fm);
  }
#endif
}

// ---------------------------------------------------------------------------
// Launch.  Output layout (floats, reference return order):
//   [0]           nms_features  100*4096 = 409600
//   [409600]      final_boxes   400
//   [410000]      final_scores  100
//   [410100]      final_labels  100
//   [410200]      valid_out     100
//   [410300]      boxes_per_cls 100*151*4 = 60400      (total 470700)
// Workspace: ~2.1 MB.
// ---------------------------------------------------------------------------
extern "C" void kernel_launch(void* const* d_in, const int* in_sizes, int n_in,
                              void* d_out, int out_size, void* d_ws, size_t ws_size,
                              hipStream_t stream)
{
  (void)in_sizes; (void)n_in; (void)out_size; (void)ws_size;
  const float* features = (const float*)d_in[0];
  const float* logits   = (const float*)d_in[1];
  const float* reg      = (const float*)d_in[2];
  const float* pboxes   = (const float*)d_in[3];
  float* out = (float*)d_out;

  // workspace carve-up
  float* probs      = (float*)d_ws;                          // 600*151
  float* boxes      = probs + N_PROP * NUM_CLASSES;          // 600*151*4
  float* sel        = boxes + N_PROP * NUM_CLASSES * 4;      // 600
  int*   kept       = (int*)(sel + N_PROP);                  // 151*600
  int*   labels_pre = kept + NUM_CLASSES * N_PROP;           // 600
  int*   top_idx    = labels_pre + N_PROP;                   // 100
  int*   top_m      = top_idx + DETS;                        // 100

  float* out_feat   = out;
  float* out_boxes  = out + 409600;
  float* out_scores = out + 410000;
  float* out_labels = out + 410100;
  float* out_valid  = out + 410200;
  float* out_bpc    = out + 410300;

  k_softmax_decode<<<(N_PROP + 3) / 4, 128, 0, stream>>>(logits, reg, pboxes,
                                                         probs, boxes);
  k_nms_per_class<<<NUM_CLASSES - 1, NMS_BLOCK, 0, stream>>>(probs, boxes, kept);
  k_label_select<<<(N_PROP + 127) / 128, 128, 0, stream>>>(probs, kept, sel,
                                                           labels_pre);
  k_topk<<<1, NMS_BLOCK, 0, stream>>>(sel, labels_pre, boxes, features,
                                      out_boxes, out_scores, out_labels,
                                      out_valid, top_idx, top_m);
  k_boxes_per_cls<<<DETS, 256, 0, stream>>>(boxes, top_idx, top_m, out_bpc);
  k_feat_gather<<<DETS, 256, 0, stream>>>(features, top_idx, top_m, out_feat);
}